// Attention_WCC_36575941492779
// MI455X (gfx1250) — compile-verified
//
#include <hip/hip_runtime.h>

// ---------------------------------------------------------------------------
// Wavelet coherence (WCT) for MI455X / gfx1250, compile-only target.
//
// One workgroup (256 threads = 8 x wave32) per batch pair. Every 2048-pt FFT
// is decomposed as 16 x 128: sixteen length-128 Stockham radix-2 FFTs in LDS
// (VALU), an inter-stage twiddle, and a radix-16 combine expressed as a
// complex 16x16 DFT GEMM on the matrix pipe: 4 real V_WMMA_F32_16X16X4_F32
// chains (K=16 -> 4 chained wmmas) x 8 column tiles (one tile per wave).
//
// All per-batch state lives in 136 KB of the 320 KB WGP LDS. The 10-deep
// scale-smoothing ring (21 MB across 64 batches) lives in a device global and
// is fully resident in the 192 MB L2. Input traffic is ~2 MB total, so the
// kernel is on-chip compute bound, which is why the DFT combine is folded
// onto the WMMA pipe (f32 WMMA keeps reference precision).
// ---------------------------------------------------------------------------

typedef __attribute__((ext_vector_type(2))) float v2f;
typedef __attribute__((ext_vector_type(8))) float v8f;

#define NPT        2048
#define NTHR       256
#define MAXB       64
#define RING_SLOTS 10
#define WSZ        94                 // WINDOW*3 - 2
#define OUTN       (NPT - WSZ + 1)    // 1955

// scale-smoothing ring: [batch][slot][quantity: S1,S2,ReS12,ImS12][time]
__device__ float g_ring[(size_t)MAXB * RING_SLOTS * 4 * NPT];

struct AFrags { v2f cr[4]; v2f sp[4]; v2f sn[4]; };

// DFT16 operand fragments for V_WMMA_F32_16X16X4_F32.
// A-matrix 16x4 f32 layout: lanes 0-15 hold rows M=0..15 with K-cols {0,1} in
// the two VGPRs; lanes 16-31 hold K-cols {2,3}. Chunk c covers K = 4c..4c+3.
// cr = cos, sp = +sin, sn = -sin  (sign of the imaginary part is selected per
// forward/inverse at the call site; no reliance on wmma NEG semantics).
__device__ __forceinline__ void make_frags(int lane, AFrags& f) {
  int m = lane & 15;
  int khalf = (lane >> 4) << 1;
#pragma unroll
  for (int c = 0; c < 4; ++c) {
    int k0 = (c << 2) + khalf;
    float s0, c0, s1, c1;
    __sincosf(0.39269908169872415f * (float)(m * k0), &s0, &c0);        // 2*pi/16
    __sincosf(0.39269908169872415f * (float)(m * (k0 + 1)), &s1, &c1);
    f.cr[c].x = c0;  f.cr[c].y = c1;
    f.sp[c].x = s0;  f.sp[c].y = s1;
    f.sn[c].x = -s0; f.sn[c].y = -s1;
  }
}

// 2048-pt complex FFT (INV=false: forward, INV=true: inverse incl. 1/N).
// in/out natural order; out may alias in (input consumed in the first step).
// war/wai/wbr/wbi: LDS ping-pong work buffers (2048 f32 each).
template<bool INV>
__device__ __forceinline__ void fft2048(const float* __restrict__ inr,
                                        const float* __restrict__ ini,
                                        float* __restrict__ outr,
                                        float* __restrict__ outi,
                                        float* war, float* wai,
                                        float* wbr, float* wbi,
                                        const AFrags& fr, int tid)
{
  const float SGN = INV ? 1.0f : -1.0f;
  // digit-swap load: A[n1*128 + n2] = in[n1 + 16*n2]
  for (int e = tid; e < NPT; e += NTHR) {
    int n1 = e >> 7, n2 = e & 127;
    int sidx = n1 + (n2 << 4);
    war[e] = inr[sidx];
    wai[e] = ini[sidx];
  }
  __syncthreads();
  float *sr = war, *si = wai, *dr = wbr, *di = wbi;
  // 16 independent 128-pt Stockham radix-2 FFTs (natural order in/out).
  // Per stage: 64 butterflies per block, stride st, st*m == 64 always.
  for (int stage = 0; stage < 7; ++stage) {
    int ncur = 128 >> stage;
    int st = 1 << stage;
    float thb = SGN * 6.283185307179586f / (float)ncur;
    for (int u = tid; u < 1024; u += NTHR) {
      int blk = (u >> 6) << 7;
      int r = u & 63;
      int p = r >> stage;
      int q = r & (st - 1);
      int ia = blk + q + st * p;
      float ar = sr[ia],      ai = si[ia];
      float br = sr[ia + 64], bi = si[ia + 64];
      float sv, cv;
      __sincosf(thb * (float)p, &sv, &cv);
      int id = blk + q + ((st * p) << 1);
      dr[id] = ar + br;  di[id] = ai + bi;
      float xr = ar - br, xi = ai - bi;
      dr[id + st] = xr * cv - xi * sv;
      di[id + st] = xr * sv + xi * cv;
    }
    __syncthreads();
    float* t;
    t = sr; sr = dr; dr = t;
    t = si; si = di; di = t;
  }
  // inter-stage twiddle: T[n1][k2] *= exp(SGN*2i*pi*n1*k2/2048)
  for (int e = tid; e < NPT; e += NTHR) {
    int n1 = e >> 7, k2 = e & 127;
    float sv, cv;
    __sincosf(SGN * 0.0030679615757712823f * (float)(n1 * k2), &sv, &cv);
    float r0 = sr[e], i0 = si[e];
    sr[e] = r0 * cv - i0 * sv;
    si[e] = r0 * sv + i0 * cv;
  }
  __syncthreads();
  // Radix-16 combine on the matrix pipe:
  //   X[128*k1 + k2] = sum_n1 W16^(SGN*k1*n1) * T[n1][k2]
  // Complex [16x16] @ [16x128] GEMM; one 16-column tile per wave32.
  // B-matrix 4x16 f32: lanes 0-15 rows K={0,1}(VGPR0/1), lanes 16-31 K={2,3}.
  int lane = tid & 31;
  int wv = tid >> 5;
  int col = (wv << 4) + (lane & 15);
  int khalf = (lane >> 4) << 1;
  v8f accr = {};
  v8f acci = {};
#pragma unroll
  for (int c = 0; c < 4; ++c) {
    int row0 = (c << 2) + khalf;
    v2f brf, bif;
    brf.x = sr[(row0 << 7) + col];
    brf.y = sr[((row0 + 1) << 7) + col];
    bif.x = si[(row0 << 7) + col];
    bif.y = si[((row0 + 1) << 7) + col];
    // D_re = Cr@Br - Im@Bi ; D_im = Cr@Bi + Im@Br ; Im = -sin (fwd), +sin (inv)
    accr = __builtin_amdgcn_wmma_f32_16x16x4_f32(false, fr.cr[c], false, brf,
                                                 (short)0, accr, false, false);
    accr = __builtin_amdgcn_wmma_f32_16x16x4_f32(false, INV ? fr.sn[c] : fr.sp[c],
                                                 false, bif, (short)0, accr, false, false);
    acci = __builtin_amdgcn_wmma_f32_16x16x4_f32(false, fr.cr[c], false, bif,
                                                 (short)0, acci, false, false);
    acci = __builtin_amdgcn_wmma_f32_16x16x4_f32(false, INV ? fr.sp[c] : fr.sn[c],
                                                 false, brf, (short)0, acci, false, false);
  }
  const float osc = INV ? (1.0f / 2048.0f) : 1.0f;
  // D 16x16 f32: VGPR v -> row M = v (+8 for upper lanes), col N = lane&15.
  int mbase = (lane >> 4) << 3;
#pragma unroll
  for (int v = 0; v < 8; ++v) {
    int k1 = mbase + v;
    outr[(k1 << 7) + col] = accr[v] * osc;
    outi[(k1 << 7) + col] = acci[v] * osc;
  }
  __syncthreads();
}

__global__ void __launch_bounds__(NTHR)
wct_wmma_kernel(const float* __restrict__ x, float* __restrict__ out)
{
  extern __shared__ float sm[];
  float* s_in_r = sm + 0 * NPT;
  float* s_in_i = sm + 1 * NPT;
  float* s_Ar   = sm + 2 * NPT;   // Stockham ping
  float* s_Ai   = sm + 3 * NPT;
  float* s_Br   = sm + 4 * NPT;   // Stockham pong
  float* s_Bi   = sm + 5 * NPT;
  float* s_y1r  = sm + 6 * NPT;   // fft(y1)
  float* s_y1i  = sm + 7 * NPT;
  float* s_y2r  = sm + 8 * NPT;   // fft(y2)
  float* s_y2i  = sm + 9 * NPT;
  float* s_W1r  = sm + 10 * NPT;  // CWT rows at current scale
  float* s_W1i  = sm + 11 * NPT;
  float* s_W2r  = sm + 12 * NPT;
  float* s_W2i  = sm + 13 * NPT;
  float* s_or   = sm + 14 * NPT;  // smoothing fft in/out
  float* s_oi   = sm + 15 * NPT;
  float* s_coh  = sm + 16 * NPT;  // coherence accumulator
  __shared__ float s_red[NTHR];

  const int b = blockIdx.x;
  const int tid = threadIdx.x;
  AFrags fr;
  make_frags(tid & 31, fr);

  // Morlet scale ladder (matches reference): s0, J = int(8*log2(n*dt/s0))
  const float s0 = 2.0f * 0.1f * (6.0f + sqrtf(38.0f)) / (4.0f * 3.14159265358979f);
  const int NS = (int)(8.0f * log2f(2048.0f * 0.1f / s0)) + 1;   // 81

  // ---- normalize inputs and take forward FFTs -----------------------------
  for (int sig = 0; sig < 2; ++sig) {
    const float* y = x + ((size_t)(b * 2 + sig)) * NPT;
    float ls = 0.f, lq = 0.f;
    for (int i = tid; i < NPT; i += NTHR) { float v = y[i]; ls += v; lq += v * v; }
    s_red[tid] = ls;
    for (int off = 128; off > 0; off >>= 1) { __syncthreads(); if (tid < off) s_red[tid] += s_red[tid + off]; }
    __syncthreads();
    float mean = s_red[0] * (1.0f / NPT);
    __syncthreads();
    s_red[tid] = lq;
    for (int off = 128; off > 0; off >>= 1) { __syncthreads(); if (tid < off) s_red[tid] += s_red[tid + off]; }
    __syncthreads();
    float var = s_red[0] * (1.0f / NPT) - mean * mean;
    float istd = rsqrtf(var);
    __syncthreads();
    for (int i = tid; i < NPT; i += NTHR) { s_in_r[i] = (y[i] - mean) * istd; s_in_i[i] = 0.f; }
    __syncthreads();
    if (sig == 0) fft2048<false>(s_in_r, s_in_i, s_y1r, s_y1i, s_Ar, s_Ai, s_Br, s_Bi, fr, tid);
    else          fft2048<false>(s_in_r, s_in_i, s_y2r, s_y2i, s_Ar, s_Ai, s_Br, s_Bi, fr, tid);
  }
  for (int i = tid; i < NPT; i += NTHR) s_coh[i] = 0.f;
  __syncthreads();

  float* ringb = g_ring + (size_t)b * RING_SLOTS * 4 * NPT;

  // ---- stream over scales; emit scale-smoothed rows with 4-scale latency --
  for (int s = 0; s < NS + 4; ++s) {
    if (s < NS) {
      float scl = s0 * exp2f(0.125f * (float)s);
      float inv_s = 1.0f / scl;
      float cnorm = sqrtf(62.83185307179586f * scl) * 0.7511255444649425f; // sqrt(2*pi*s/dt)*pi^-0.25
      float sdt = scl * 10.0f;                                             // s/dt

      // CWT: W = ifft(yh * norm * psi_hat), psi_hat only on w>0 (k=1..1023)
      for (int k = tid; k < NPT; k += NTHR) {
        float h = 0.f;
        if (k >= 1 && k < 1024) {
          float w = 6.283185307179586f * (float)k * (1.0f / 204.8f);
          float u = scl * w - 6.0f;
          h = cnorm * __expf(-0.5f * u * u);
        }
        s_in_r[k] = s_y1r[k] * h;
        s_in_i[k] = s_y1i[k] * h;
      }
      __syncthreads();
      fft2048<true>(s_in_r, s_in_i, s_W1r, s_W1i, s_Ar, s_Ai, s_Br, s_Bi, fr, tid);
      for (int k = tid; k < NPT; k += NTHR) {
        float h = 0.f;
        if (k >= 1 && k < 1024) {
          float w = 6.283185307179586f * (float)k * (1.0f / 204.8f);
          float u = scl * w - 6.0f;
          h = cnorm * __expf(-0.5f * u * u);
        }
        s_in_r[k] = s_y2r[k] * h;
        s_in_i[k] = s_y2i[k] * h;
      }
      __syncthreads();
      fft2048<true>(s_in_r, s_in_i, s_W2r, s_W2i, s_Ar, s_Ai, s_Br, s_Bi, fr, tid);

      // time-smooth P1 and P2 packed as one complex row (filter is real-even)
      for (int k = tid; k < NPT; k += NTHR) {
        s_in_r[k] = (s_W1r[k] * s_W1r[k] + s_W1i[k] * s_W1i[k]) * inv_s;
        s_in_i[k] = (s_W2r[k] * s_W2r[k] + s_W2i[k] * s_W2i[k]) * inv_s;
      }
      __syncthreads();
      fft2048<false>(s_in_r, s_in_i, s_or, s_oi, s_Ar, s_Ai, s_Br, s_Bi, fr, tid);
      for (int k = tid; k < NPT; k += NTHR) {
        float f = (float)(k < 1024 ? k : k - 2048) * (1.0f / 2048.0f);
        float kw = 6.283185307179586f * f;
        float F = __expf(-0.5f * sdt * sdt * kw * kw);
        s_or[k] *= F; s_oi[k] *= F;
      }
      __syncthreads();
      fft2048<true>(s_or, s_oi, s_or, s_oi, s_Ar, s_Ai, s_Br, s_Bi, fr, tid);
      {
        float* rp = ringb + (size_t)(s % RING_SLOTS) * 4 * NPT;
        for (int t = tid; t < NPT; t += NTHR) { rp[t] = s_or[t]; rp[NPT + t] = s_oi[t]; }
      }

      // time-smooth cross spectrum W1*conj(W2)/s (complex)
      for (int k = tid; k < NPT; k += NTHR) {
        s_in_r[k] = (s_W1r[k] * s_W2r[k] + s_W1i[k] * s_W2i[k]) * inv_s;
        s_in_i[k] = (s_W1i[k] * s_W2r[k] - s_W1r[k] * s_W2i[k]) * inv_s;
      }
      __syncthreads();
      fft2048<false>(s_in_r, s_in_i, s_or, s_oi, s_Ar, s_Ai, s_Br, s_Bi, fr, tid);
      for (int k = tid; k < NPT; k += NTHR) {
        float f = (float)(k < 1024 ? k : k - 2048) * (1.0f / 2048.0f);
        float kw = 6.283185307179586f * f;
        float F = __expf(-0.5f * sdt * sdt * kw * kw);
        s_or[k] *= F; s_oi[k] *= F;
      }
      __syncthreads();
      fft2048<true>(s_or, s_oi, s_or, s_oi, s_Ar, s_Ai, s_Br, s_Bi, fr, tid);
      {
        float* rp = ringb + (size_t)(s % RING_SLOTS) * 4 * NPT;
        for (int t = tid; t < NPT; t += NTHR) { rp[2 * NPT + t] = s_or[t]; rp[3 * NPT + t] = s_oi[t]; }
      }
    }

    // scale-smoothing ('same' conv, 10-tap rect with half-weight ends) and
    // coherence accumulation for row r = s - 4 (needs rows r-5..r+4, all of
    // which are resident in the ring: slots hold rows s-9..s).
    int r = s - 4;
    if (r >= 0 && r < NS) {
      __syncthreads();
      for (int t = tid; t < NPT; t += NTHR) {
        float a1 = 0.f, a2 = 0.f, cr = 0.f, ci = 0.f;
#pragma unroll
        for (int i = 0; i < 10; ++i) {
          int row = r + 4 - i;
          if (row >= 0 && row < NS) {
            float w = (i == 0 || i == 9) ? (0.5f / 9.0f) : (1.0f / 9.0f);
            const float* rq = ringb + (size_t)(row % RING_SLOTS) * 4 * NPT;
            a1 += w * rq[t];
            a2 += w * rq[NPT + t];
            cr += w * rq[2 * NPT + t];
            ci += w * rq[3 * NPT + t];
          }
        }
        s_coh[t] += (cr * cr + ci * ci) / (a1 * a2);
      }
      __syncthreads();
    }
  }
  __syncthreads();

  // sliding-window sum over WS=94 -> [OUTN] per batch
  for (int i = tid; i < OUTN; i += NTHR) {
    float acc = 0.f;
    for (int j = 0; j < WSZ; ++j) acc += s_coh[i + j];
    out[(size_t)b * OUTN + i] = acc;
  }
}

extern "C" void kernel_launch(void* const* d_in, const int* in_sizes, int n_in,
                              void* d_out, int out_size, void* d_ws, size_t ws_size,
                              hipStream_t stream) {
  (void)n_in; (void)out_size; (void)d_ws; (void)ws_size;
  const float* x = (const float*)d_in[0];
  float* out = (float*)d_out;
  int B = in_sizes[0] / (2 * NPT);     // 64
  if (B > MAXB) B = MAXB;
  size_t shmem = (size_t)17 * NPT * sizeof(float);   // 136 KB of 320 KB WGP LDS
  wct_wmma_kernel<<<dim3(B), dim3(NTHR), shmem, stream>>>(x, out);
}